// GraphAttentionLayer_22230750724383
// MI455X (gfx1250) — compile-verified
//
#include <hip/hip_runtime.h>

typedef __attribute__((ext_vector_type(2))) float v2f;
typedef __attribute__((ext_vector_type(4))) float v4f;
typedef __attribute__((ext_vector_type(8))) float v8f;

#define IN_F   128
#define OUT_F  64
#define ALPHA  0.2f
#define INV_SQRT_F 0.125f   /* 1/sqrt(OUT_F) = 1/8 */

// ---------------------------------------------------------------------------
// Kernel 1: data[N][64] = h[N][128] @ W[64][128]^T + bias, via fp32 WMMA.
// One wave per 16x16 output tile; K stepped by 4 (V_WMMA_F32_16X16X4_F32).
// A-layout (16x4 f32): lane L holds row M=L%16; VGPR0/1 = K = 2*(L/16)+{0,1}.
// B-layout (4x16 f32): lane L holds col N=L%16; VGPR i  = K = i + 2*(L/16).
// Both are contiguous float2 loads from row-major h / W.
// C/D layout: VGPR r, lane L -> element (M = r + 8*(L/16), N = L%16).
//
// Out-of-range rows: A row M only feeds D row M, and those D rows are never
// stored — so boundary lanes just load a clamped (row N-1) address; no zeroing
// needed, keeping the hot loop pure load+wmma.
// ---------------------------------------------------------------------------
__global__ __launch_bounds__(256) void gat_gemm_wmma(
    const float* __restrict__ h,     // [N][IN_F]
    const float* __restrict__ W,     // [OUT_F][IN_F]
    const float* __restrict__ bias,  // [OUT_F]
    float* __restrict__ data,        // [N][OUT_F]
    int N, int numTiles)
{
  int wave = (int)((blockIdx.x * blockDim.x + threadIdx.x) >> 5);
  if (wave >= numTiles) return;               // wave-uniform exit, EXEC stays all-1

  int rowTile = wave >> 2;                    // OUT_F/16 = 4 column tiles
  int colTile = wave & 3;
  int lane = threadIdx.x & 31;
  int half = lane >> 4;                       // 0: lanes 0-15, 1: lanes 16-31
  int l16  = lane & 15;
  int row0 = rowTile << 4;
  int col0 = colTile << 4;

  int m = row0 + l16;
  int mclamp = (m < N) ? m : (N - 1);         // clamped: garbage rows never stored
  const float* hrow = h + (size_t)mclamp * IN_F;
  const float* wrow = W + (size_t)(col0 + l16) * IN_F;

  // C initialized with bias broadcast along M -> fuses "+ W_b"
  float bv = bias[col0 + l16];
  v8f c;
  #pragma unroll
  for (int r = 0; r < 8; ++r) c[r] = bv;

  #pragma unroll
  for (int k = 0; k < IN_F; k += 4) {
    v2f a = *(const v2f*)(hrow + k + 2 * half);
    v2f b = *(const v2f*)(wrow + k + 2 * half);
    // (neg_a, A, neg_b, B, c_mod, C, reuse_a, reuse_b)
    c = __builtin_amdgcn_wmma_f32_16x16x4_f32(false, a, false, b,
                                              (short)0, c, false, false);
  }

  #pragma unroll
  for (int r = 0; r < 8; ++r) {
    int mm = row0 + r + 8 * half;
    if (mm < N) data[(size_t)mm * OUT_F + col0 + l16] = c[r];
  }
}

// ---------------------------------------------------------------------------
// Kernel 2: per-node attention scores s_src = data.a_src, s_dst = data.a_dst
// ---------------------------------------------------------------------------
__global__ __launch_bounds__(256) void gat_scores(
    const float* __restrict__ data, const float* __restrict__ a,
    float* __restrict__ s_src, float* __restrict__ s_dst, int N)
{
  int i = (int)(blockIdx.x * blockDim.x + threadIdx.x);
  if (i >= N) return;
  const float* row = data + (size_t)i * OUT_F;
  float ss = 0.0f, sd = 0.0f;
  #pragma unroll
  for (int f = 0; f < OUT_F; f += 4) {
    v4f v  = *(const v4f*)(row + f);
    v4f as = *(const v4f*)(a + f);
    v4f ad = *(const v4f*)(a + OUT_F + f);
    ss = fmaf(v[0], as[0], fmaf(v[1], as[1], fmaf(v[2], as[2], fmaf(v[3], as[3], ss))));
    sd = fmaf(v[0], ad[0], fmaf(v[1], ad[1], fmaf(v[2], ad[2], fmaf(v[3], ad[3], sd))));
  }
  s_src[i] = ss;
  s_dst[i] = sd;
}

// ---------------------------------------------------------------------------
// Kernel 3: edge scatter. 16 threads per edge, float4 of features per thread.
// edge_e = exp(leaky_relu(s_src[src]+s_dst[dst]) / 8)
// out[src][f..f+3] += edge_e * data[dst][f..f+3]  (b128 gather + 4 f32 atomics)
// rowsum[src] += edge_e                           (first thread of edge group)
// ---------------------------------------------------------------------------
__global__ __launch_bounds__(256) void gat_edges(
    const int* __restrict__ src, const int* __restrict__ dst,
    const float* __restrict__ s_src, const float* __restrict__ s_dst,
    const float* __restrict__ data, float* __restrict__ out,
    float* __restrict__ rowsum, long long E)
{
  long long t = (long long)blockIdx.x * blockDim.x + threadIdx.x;
  long long e = t >> 4;                       // 16 threads per edge
  if (e >= E) return;
  int f = (int)(t & 15) * 4;                  // feature quad

  int s = src[e];
  int d = dst[e];
  float sc = s_src[s] + s_dst[d];
  sc = (sc > 0.0f) ? sc : ALPHA * sc;
  float ee = __expf(sc * INV_SQRT_F);

  v4f dv = *(const v4f*)(data + (size_t)d * OUT_F + f);
  float* op = out + (size_t)s * OUT_F + f;
  unsafeAtomicAdd(op + 0, ee * dv[0]);
  unsafeAtomicAdd(op + 1, ee * dv[1]);
  unsafeAtomicAdd(op + 2, ee * dv[2]);
  unsafeAtomicAdd(op + 3, ee * dv[3]);
  if (f == 0) unsafeAtomicAdd(&rowsum[s], ee);
}

// ---------------------------------------------------------------------------
// Kernel 4: finalize. rowsum==0 rows fall back to data[i], divisor 1.
// One float4 per thread.
// ---------------------------------------------------------------------------
__global__ __launch_bounds__(256) void gat_finalize(
    float* __restrict__ out, const float* __restrict__ data,
    const float* __restrict__ rowsum, long long totalQuads)
{
  long long q = (long long)blockIdx.x * blockDim.x + threadIdx.x;
  if (q >= totalQuads) return;
  long long i = q >> 4;                       // node index (OUT_F/4 = 16 quads/row)
  float rs = rowsum[i];
  v4f v = *(v4f*)(out + q * 4);
  if (rs == 0.0f) {                           // accumulated row is exactly 0 here
    v = *(const v4f*)(data + q * 4);
    rs = 1.0f;
  }
  float inv = 1.0f / rs;
  v[0] *= inv; v[1] *= inv; v[2] *= inv; v[3] *= inv;
  *(v4f*)(out + q * 4) = v;
}

// ---------------------------------------------------------------------------
extern "C" void kernel_launch(void* const* d_in, const int* in_sizes, int n_in,
                              void* d_out, int out_size, void* d_ws, size_t ws_size,
                              hipStream_t stream)
{
  const float* h    = (const float*)d_in[0];   // [N][128]
  const int*   adj  = (const int*)d_in[1];     // [2][E]
  const float* W    = (const float*)d_in[2];   // [64][128]
  const float* bias = (const float*)d_in[3];   // [64]
  const float* a    = (const float*)d_in[4];   // [1][128]
  float* out = (float*)d_out;                  // [N][64]

  int       N = in_sizes[0] / IN_F;
  long long E = in_sizes[1] / 2;

  // Workspace layout: data[N][64] | s_src[N] | s_dst[N] | rowsum[N]
  float* data   = (float*)d_ws;
  float* s_src  = data + (size_t)N * OUT_F;
  float* s_dst  = s_src + N;
  float* rowsum = s_dst + N;

  hipMemsetAsync(out,    0, (size_t)out_size * sizeof(float), stream);
  hipMemsetAsync(rowsum, 0, (size_t)N * sizeof(float), stream);

  int numTiles = ((N + 15) / 16) * (OUT_F / 16);
  int gemmBlocks = (numTiles * 32 + 255) / 256;     // 8 wave-tiles per block
  gat_gemm_wmma<<<gemmBlocks, 256, 0, stream>>>(h, W, bias, data, N, numTiles);

  gat_scores<<<(N + 255) / 256, 256, 0, stream>>>(data, a, s_src, s_dst, N);

  long long edgeThreads = E * 16;
  gat_edges<<<(unsigned)((edgeThreads + 255) / 256), 256, 0, stream>>>(
      adj, adj + E, s_src, s_dst, data, out, rowsum, E);

  long long totalQuads = (long long)N * (OUT_F / 4);
  gat_finalize<<<(unsigned)((totalQuads + 255) / 256), 256, 0, stream>>>(
      out, data, rowsum, totalQuads);
}